// SelfMixing_79886391705983
// MI455X (gfx1250) — compile-verified
//
#include <hip/hip_runtime.h>
#include <hip/hip_bf16.h>

typedef float v2f __attribute__((ext_vector_type(2)));
typedef float v8f __attribute__((ext_vector_type(8)));

static constexpr int FEAT    = 256;   // features
static constexpr int P_PAD   = 208;   // 207 z-entries padded to 52 chunks of K=4
static constexpr int NCHUNKS = 52;
static constexpr int ZSTRIDE = 212;   // LDS row stride (floats) to avoid bank conflicts
static constexpr int KXSTRIDE = 33;

// ---------------------------------------------------------------------------
// Setup kernel: pack the 11 CG tensors into a dense A matrix (16 x 208),
// row r = L*L + k (matching output concat layout), column = z index.
// ---------------------------------------------------------------------------
__global__ void pack_A_kernel(const float* __restrict__ cg0,  const float* __restrict__ cg1,
                              const float* __restrict__ cg2,  const float* __restrict__ cg3,
                              const float* __restrict__ cg4,  const float* __restrict__ cg5,
                              const float* __restrict__ cg6,  const float* __restrict__ cg7,
                              const float* __restrict__ cg8,  const float* __restrict__ cg9,
                              const float* __restrict__ cg10, float* __restrict__ A) {
  const int t = threadIdx.x;
  for (int i = t; i < 16 * P_PAD; i += blockDim.x) A[i] = 0.0f;
  __syncthreads();
#define FILL_CG(CG, NIJ, NL, KB, BASE)                                  \
  for (int idx = t; idx < (NIJ) * (NL); idx += blockDim.x) {            \
    int k  = idx % (NL);                                                \
    int ij = idx / (NL);                                                \
    A[((KB) + k) * P_PAD + (BASE) + ij] = (CG)[idx];                    \
  }
  //        cg    (2l1+1)(2l2+1)  2L+1  rowbase  zbase
  FILL_CG(cg0,        1*3,         3,     1,       0)   // (0,1,1)
  FILL_CG(cg1,        1*5,         5,     4,       3)   // (0,2,2)
  FILL_CG(cg2,        1*7,         7,     9,       8)   // (0,3,3)
  FILL_CG(cg3,        3*5,         3,     1,      15)   // (1,2,1)
  FILL_CG(cg4,        3*5,         5,     4,      30)   // (1,2,2)
  FILL_CG(cg5,        3*5,         7,     9,      45)   // (1,2,3)
  FILL_CG(cg6,        3*7,         5,     4,      60)   // (1,3,2)
  FILL_CG(cg7,        3*7,         7,     9,      81)   // (1,3,3)
  FILL_CG(cg8,        5*7,         3,     1,     102)   // (2,3,1)
  FILL_CG(cg9,        5*7,         5,     4,     137)   // (2,3,2)
  FILL_CG(cg10,       5*7,         7,     9,     172)   // (2,3,3)
#undef FILL_CG
}

// ---------------------------------------------------------------------------
// Main kernel: one wave per block. Block handles one 32-feature tile and a
// strided range of nodes n. Per n: lanes build z rows in LDS, then two
// 16-slot WMMA tiles of 52 chained V_WMMA_F32_16X16X4_F32 each.
// ---------------------------------------------------------------------------
struct MixArgs {
  const float* x0; const float* x1; const float* x2; const float* x3;
  const float* keep0; const float* keep1; const float* keep2; const float* keep3;
  const float* mix[11];
  const float* Apk;
  float* out;
  int N;
};

__global__ __launch_bounds__(32) void selfmix_wmma_kernel(MixArgs args) {
  __shared__ float zsh[32 * ZSTRIDE];    // z rows, slot-major
  __shared__ float kxsh[16 * KXSTRIDE];  // keep*x, comp-major

  const int lane = threadIdx.x;          // 0..31
  const int half = lane >> 4;            // 0 or 1
  const int col  = lane & 15;
  const int ft   = blockIdx.x & 7;       // 8 feature tiles of 32
  const int f0   = ft * 32;
  const int f    = f0 + lane;
  const int nstart  = blockIdx.x >> 3;
  const int nstride = gridDim.x >> 3;
  const int N = args.N;
  const long NF = (long)N * FEAT;

  // per-feature coefficients (resident in regs for the whole n-loop)
  float mixv[11];
#pragma unroll
  for (int c = 0; c < 11; ++c) mixv[c] = args.mix[c][f];
  float keepv[4] = { args.keep0[f], args.keep1[f], args.keep2[f], args.keep3[f] };

  // A-matrix fragments: lane holds row=col, K = 4k + 2*half + {0,1}
  v2f a[NCHUNKS];
  {
    const float* ab = args.Apk + col * P_PAD + 2 * half;
#pragma unroll
    for (int k = 0; k < NCHUNKS; ++k) a[k] = *(const v2f*)(ab + 4 * k);
  }

  // per-lane output addressing: rows r = v + 8*half; y_L starts at L*L*N*F
  long compOff[8]; int rowMul[8];
#pragma unroll
  for (int v = 0; v < 8; ++v) {
    const int r = v + 8 * half;
    const int L = (r == 0) ? 0 : (r < 4 ? 1 : (r < 9 ? 2 : 3));
    const int kk = r - L * L;
    compOff[v] = (long)(L * L) * NF + (long)kk * FEAT;
    rowMul[v]  = (2 * L + 1) * FEAT;
  }

  for (int n = nstart; n < N; n += nstride) {
    // ---- load the 16 irrep components for this lane's (n, f) slot ----
    float xc[16];
    xc[0] = args.x0[(long)n * FEAT + f];
#pragma unroll
    for (int i = 0; i < 3; ++i) xc[1 + i] = args.x1[((long)n * 3 + i) * FEAT + f];
#pragma unroll
    for (int i = 0; i < 5; ++i) xc[4 + i] = args.x2[((long)n * 5 + i) * FEAT + f];
#pragma unroll
    for (int i = 0; i < 7; ++i) xc[9 + i] = args.x3[((long)n * 7 + i) * FEAT + f];

    // ---- keep path: keep_L[f] * x into LDS (becomes the C accumulator) ----
#pragma unroll
    for (int c2 = 0; c2 < 16; ++c2) {
      const int L = (c2 == 0) ? 0 : (c2 < 4 ? 1 : (c2 < 9 ? 2 : 3));
      kxsh[c2 * KXSTRIDE + lane] = keepv[L] * xc[c2];
    }

    // ---- z row: mix-scaled outer products, 207 entries ----
    float* zr = &zsh[lane * ZSTRIDE];
#define DO_COMBO(CI, O1, N1, O2, N2, BASE)                              \
    { _Pragma("unroll") for (int i = 0; i < (N1); ++i) {                \
        const float t = mixv[CI] * xc[(O1) + i];                        \
        _Pragma("unroll") for (int j = 0; j < (N2); ++j)                \
          zr[(BASE) + i * (N2) + j] = t * xc[(O2) + j];                 \
      } }
    DO_COMBO(0,  0, 1, 1, 3,   0)   // (0,1,1)
    DO_COMBO(1,  0, 1, 4, 5,   3)   // (0,2,2)
    DO_COMBO(2,  0, 1, 9, 7,   8)   // (0,3,3)
    DO_COMBO(3,  1, 3, 4, 5,  15)   // (1,2,1)
    DO_COMBO(4,  1, 3, 4, 5,  30)   // (1,2,2)
    DO_COMBO(5,  1, 3, 4, 5,  45)   // (1,2,3)
    DO_COMBO(6,  1, 3, 9, 7,  60)   // (1,3,2)
    DO_COMBO(7,  1, 3, 9, 7,  81)   // (1,3,3)
    DO_COMBO(8,  4, 5, 9, 7, 102)   // (2,3,1)
    DO_COMBO(9,  4, 5, 9, 7, 137)   // (2,3,2)
    DO_COMBO(10, 4, 5, 9, 7, 172)   // (2,3,3)
#undef DO_COMBO
    zr[207] = 0.0f;                 // pad (A column 207 is zero too)
    __syncthreads();                // single wave: acts as DS fence

    // ---- two 16-slot WMMA tiles ----
#pragma unroll
    for (int tile = 0; tile < 2; ++tile) {
      const int tb = tile * 16;
      // C init = keep path (lane holds rows v+8*half, column `col`)
      v8f c = {};
#pragma unroll
      for (int v = 0; v < 8; ++v)
        c[v] = kxsh[(v + 8 * half) * KXSTRIDE + tb + col];

      // chained WMMA over 52 K-chunks; B lane = (row 4k+2*half+{0,1}, col slot)
      const float* zb = &zsh[(tb + col) * ZSTRIDE + 2 * half];
#pragma unroll
      for (int k = 0; k < NCHUNKS; ++k) {
        v2f b = *(const v2f*)(zb + 4 * k);
        c = __builtin_amdgcn_wmma_f32_16x16x4_f32(
            /*neg_a=*/false, a[k], /*neg_b=*/false, b,
            /*c_mod=*/(short)0, c, /*reuse_a=*/false, /*reuse_b=*/false);
      }

      // store D: element (r = v+8*half, feature f0+tb+col)
      const int fout = f0 + tb + col;
#pragma unroll
      for (int v = 0; v < 8; ++v)
        args.out[compOff[v] + (long)n * rowMul[v] + fout] = c[v];
    }
    __syncthreads();                // protect zsh WAR before next iteration
  }
}

// ---------------------------------------------------------------------------
extern "C" void kernel_launch(void* const* d_in, const int* in_sizes, int n_in,
                              void* d_out, int out_size, void* d_ws, size_t ws_size,
                              hipStream_t stream) {
  (void)n_in; (void)out_size; (void)ws_size;
  const float* cg[11];
  MixArgs args;
  args.x0 = (const float*)d_in[0];
  args.x1 = (const float*)d_in[1];
  args.x2 = (const float*)d_in[2];
  args.x3 = (const float*)d_in[3];
  args.keep0 = (const float*)d_in[4];
  args.keep1 = (const float*)d_in[5];
  args.keep2 = (const float*)d_in[6];
  args.keep3 = (const float*)d_in[7];
  for (int c = 0; c < 11; ++c) {
    args.mix[c] = (const float*)d_in[8 + 2 * c];
    cg[c]       = (const float*)d_in[9 + 2 * c];
  }
  float* Apk = (float*)d_ws;
  args.Apk = Apk;
  args.out = (float*)d_out;
  args.N   = in_sizes[0] / FEAT;   // x0 is (N, 1, F)

  pack_A_kernel<<<dim3(1), dim3(256), 0, stream>>>(
      cg[0], cg[1], cg[2], cg[3], cg[4], cg[5], cg[6], cg[7], cg[8], cg[9], cg[10], Apk);

  const int nblk = 1250;           // ~16 nodes per block at N=20000
  selfmix_wmma_kernel<<<dim3(8 * nblk), dim3(32), 0, stream>>>(args);
}